// Force_18726057410563
// MI455X (gfx1250) — compile-verified
//
#include <hip/hip_runtime.h>
#include <math.h>

#define N_V   2048
#define N_E   1024
#define N_MOL 64
#define V_DIM 128
#define E_DIM 64
#define H_DIM 32

typedef float v2f __attribute__((ext_vector_type(2)));
typedef float v8f __attribute__((ext_vector_type(8)));

// ---------------------------------------------------------------------------
// Recover gather indices from the one-hot matrices (coalesced single pass).
// vew1[i*N_E+j]==1 -> src[j]=i ; vew2 -> dst ; mvw[mm*N_V+i]==1 -> mol[i]=mm
// ---------------------------------------------------------------------------
__global__ void k_idx(const float* __restrict__ vew1, const float* __restrict__ vew2,
                      const float* __restrict__ mvw,
                      int* __restrict__ src, int* __restrict__ dst, int* __restrict__ mol) {
  long tid = (long)blockIdx.x * blockDim.x + threadIdx.x;
  const long n1 = (long)N_V * N_E;
  if (tid < n1) {
    if (vew1[tid] > 0.5f) { int i = (int)(tid / N_E), j = (int)(tid % N_E); src[j] = i; }
  } else if (tid < 2 * n1) {
    long t = tid - n1;
    if (vew2[t] > 0.5f) { int i = (int)(t / N_E), j = (int)(t % N_E); dst[j] = i; }
  } else if (tid < 2 * n1 + (long)N_MOL * N_V) {
    long t = tid - 2 * n1;
    if (mvw[t] > 0.5f) { int mm = (int)(t / N_V), i = (int)(t % N_V); mol[i] = mm; }
  }
}

// Deterministic per-molecule vertex lists (no atomics -> replay-stable FP order)
__global__ void k_list(const int* __restrict__ mol, int* __restrict__ molcnt,
                       unsigned short* __restrict__ mollist) {
  int mm = threadIdx.x;
  int c = 0;
  for (int i = 0; i < N_V; ++i)
    if (mol[i] == mm) mollist[mm * N_V + (c++)] = (unsigned short)i;
  molcnt[mm] = c;
}

// r = q @ Wr.T  ([2048,3] x [3,32] -> [2048,32])
__global__ void k_r(const float* __restrict__ q, const float* __restrict__ Wr,
                    float* __restrict__ r) {
  int tid = blockIdx.x * blockDim.x + threadIdx.x;   // 65536 threads
  int i = tid >> 5, h = tid & 31;
  float s = 0.f;
  #pragma unroll
  for (int d = 0; d < 3; ++d) s += q[i * 3 + d] * Wr[h * 3 + d];
  r[tid] = s;                                        // tid == i*32+h
}

// ---------------------------------------------------------------------------
// Bond MLP with f32 WMMA: one wave handles 16 directed edges.
// x[16][320] staged in LDS; C(16x16) = A(16x4) x B(4x16) + C chained over K.
// A layout: lane<16: M=lane,K=kk+{0,1}; lane>=16: M=lane-16,K=kk+{2,3}
// B layout: lane<16: N=lane,K=kk+{0,1}; lane>=16: N=lane-16,K=kk+{2,3}
// C layout: c[rr]: lane<16 -> (M=rr, N=lane); lane>=16 -> (M=rr+8, N=lane-16)
// ---------------------------------------------------------------------------
__global__ void k_bond(const float* __restrict__ v, const float* __restrict__ e,
                       const float* __restrict__ q,
                       const int* __restrict__ src, const int* __restrict__ dst,
                       const float* __restrict__ W1, const float* __restrict__ W2,
                       const float* __restrict__ b2, float* __restrict__ fbond) {
  __shared__ float xs[16][321];   // padded: bank-conflict-free
  __shared__ float hb[16][33];
  const int lane = threadIdx.x;
  const int tile = blockIdx.x;

  // cooperative gather of the 16x320 feature tile: [v[a] | e | v[b]]
  for (int idx = lane; idx < 16 * 320; idx += 32) {
    int row = idx / 320, col = idx - row * 320;
    int k = tile * 16 + row;
    int a, b, er;
    if (k < N_E) { a = src[k]; b = dst[k]; er = k; }
    else         { er = k - N_E; a = dst[er]; b = src[er]; }
    float val;
    if (col < V_DIM)              val = v[a * V_DIM + col];
    else if (col < V_DIM + E_DIM) val = e[er * E_DIM + (col - V_DIM)];
    else                          val = v[b * V_DIM + (col - V_DIM - E_DIM)];
    xs[row][col] = val;
  }
  __syncthreads();

  const int mrow  = lane & 15;
  const int khalf = (lane >> 4) << 1;       // 0 or 2
  const int moff  = (lane >> 4) << 3;       // 0 or 8

  for (int nt = 0; nt < 2; ++nt) {          // H_DIM = 32 -> two 16-wide N tiles
    const int hbase = nt * 16;
    v8f c = {0.f, 0.f, 0.f, 0.f, 0.f, 0.f, 0.f, 0.f};
    for (int kk = 0; kk < 320; kk += 4) {
      v2f a, b;
      a.x = xs[mrow][kk + khalf];
      a.y = xs[mrow][kk + khalf + 1];
      const float* wrow = W1 + (hbase + mrow) * 320 + kk + khalf;  // B = W1.T
      b.x = wrow[0];
      b.y = wrow[1];
      c = __builtin_amdgcn_wmma_f32_16x16x4_f32(false, a, false, b,
                                                (short)0, c, false, false);
    }
    #pragma unroll
    for (int rr = 0; rr < 8; ++rr)
      hb[rr + moff][hbase + mrow] = fmaxf(c[rr], 0.f);   // ReLU
  }
  __syncthreads();

  if (lane < 16) {
    int k = tile * 16 + lane;
    float val = b2[0];
    #pragma unroll
    for (int h = 0; h < H_DIM; ++h) val += hb[lane][h] * W2[h];
    int a, b;
    if (k < N_E) { a = src[k]; b = dst[k]; }
    else         { a = dst[k - N_E]; b = src[k - N_E]; }
    float dx = q[a * 3 + 0] - q[b * 3 + 0];
    float dy = q[a * 3 + 1] - q[b * 3 + 1];
    float dz = q[a * 3 + 2] - q[b * 3 + 2];
    float inv = 1.0f / sqrtf(dx * dx + dy * dy + dz * dz);
    fbond[k * 3 + 0] = dx * inv * val;
    fbond[k * 3 + 1] = dy * inv * val;
    fbond[k * 3 + 2] = dz * inv * val;
  }
}

// ---------------------------------------------------------------------------
// Relational forces: 1 wave per vertex i, lanes = 32 hidden dims.
// Only iterate j within i's molecule (mask sparsity: ~64x work reduction).
// ---------------------------------------------------------------------------
__global__ void k_rela(const float* __restrict__ r, const float* __restrict__ q,
                       const float* __restrict__ m, const float* __restrict__ br,
                       const int* __restrict__ mol, const int* __restrict__ molcnt,
                       const unsigned short* __restrict__ mollist,
                       const float* __restrict__ fbond, float* __restrict__ out) {
  const int wave = threadIdx.x >> 5;
  const int lane = threadIdx.x & 31;
  const int i = blockIdx.x * 8 + wave;

  const float ri  = r[i * 32 + lane];
  const float brh = br[lane];
  const float qi0 = q[i * 3 + 0], qi1 = q[i * 3 + 1], qi2 = q[i * 3 + 2];
  const float mi  = m[i];
  const int   mol_i = mol[i];
  const int   cnt   = molcnt[mol_i];
  const unsigned short* lst = mollist + mol_i * N_V;

  float fx = 0.f, fy = 0.f, fz = 0.f;
  for (int t = 0; t < cnt; ++t) {            // wave-uniform loop
    int j = (int)lst[t];
    if (j == i) continue;                    // uniform branch; diagonal is 0
    float d  = ri - r[j * 32 + lane] + brh;  // coalesced 128B load
    float sp = fmaxf(d, 0.f) + log1pf(__expf(-fabsf(d)));  // stable softplus
    float acc = sp * sp;
    #pragma unroll
    for (int o = 16; o > 0; o >>= 1) acc += __shfl_xor(acc, o, 32);
    // acc = delta_d^2 on every lane
    float val = (1.0f / acc - 1.0f / sqrtf(acc)) * mi * m[j];
    float dx = qi0 - q[j * 3 + 0];
    float dy = qi1 - q[j * 3 + 1];
    float dz = qi2 - q[j * 3 + 2];
    float n2 = dx * dx + dy * dy + dz * dz;
    float s  = (n2 > 0.f) ? val / sqrtf(n2) : 0.f;
    fx += dx * s; fy += dy * s; fz += dz * s;
  }
  if (lane == 0) {
    out[i * 3 + 0] = fbond[i * 3 + 0] + fx;
    out[i * 3 + 1] = fbond[i * 3 + 1] + fy;
    out[i * 3 + 2] = fbond[i * 3 + 2] + fz;
  }
}

// ---------------------------------------------------------------------------
extern "C" void kernel_launch(void* const* d_in, const int* in_sizes, int n_in,
                              void* d_out, int out_size, void* d_ws, size_t ws_size,
                              hipStream_t stream) {
  const float* v    = (const float*)d_in[0];
  const float* e    = (const float*)d_in[1];
  const float* m    = (const float*)d_in[2];
  const float* q    = (const float*)d_in[3];
  const float* vew1 = (const float*)d_in[4];
  const float* vew2 = (const float*)d_in[5];
  const float* mvw  = (const float*)d_in[6];
  const float* W1   = (const float*)d_in[7];
  const float* W2   = (const float*)d_in[8];
  const float* b2   = (const float*)d_in[9];
  const float* Wr   = (const float*)d_in[10];
  const float* br   = (const float*)d_in[11];

  char* ws = (char*)d_ws;
  int*            srcI    = (int*)(ws + 0);                 //  4 KB
  int*            dstI    = (int*)(ws + 4096);              //  4 KB
  int*            molI    = (int*)(ws + 8192);              //  8 KB
  int*            molcnt  = (int*)(ws + 16384);             // 256 B
  unsigned short* mollist = (unsigned short*)(ws + 16640);  // 256 KB
  float*          rbuf    = (float*)(ws + 278784);          // 256 KB
  float*          fbond   = (float*)(ws + 540928);          //  24 KB -> 565504 total

  long total = 2L * N_V * N_E + (long)N_MOL * N_V;
  int blocks = (int)((total + 255) / 256);
  k_idx <<<blocks, 256, 0, stream>>>(vew1, vew2, mvw, srcI, dstI, molI);
  k_list<<<1, 64, 0, stream>>>(molI, molcnt, mollist);
  k_r   <<<(N_V * 32) / 256, 256, 0, stream>>>(q, Wr, rbuf);
  k_bond<<<(2 * N_E) / 16, 32, 0, stream>>>(v, e, q, srcI, dstI, W1, W2, b2, fbond);
  k_rela<<<N_V / 8, 256, 0, stream>>>(rbuf, q, m, br, molI, molcnt, mollist, fbond,
                                      (float*)d_out);
}